// MeanAggregator_ml_90271622627848
// MI455X (gfx1250) — compile-verified
//
#include <hip/hip_runtime.h>
#include <cstdint>

// MI455X / gfx1250, wave32. Memory-bound multi-gather + mean:
//   out0[b] = (1/K) sum_k F[ni[b,k]]        out2[b] = F[nodes[b]]
//   out1[b] = (1/K) sum_k F[perm[ni[b,k]]]  out3[b] = F[perm[nodes[b]]]
//
// One wave per batch element; lane L owns bytes [16L,16L+16) of each 512B
// feature row, so every gather/store is a coalesced b128.
//
// The 20 mean-contributing rows per batch element go through the CDNA5
// async global->LDS gather path (ASYNCcnt) in SADDR form: the row base
// (features + idx*512) is wave-uniform (v_readlane), so it lives in an
// SGPR pair and a single reused lane*16 VGPR is the vector offset —
// address math stays in SALU. The 2 skip rows are plain b128 loads
// straight to VGPRs (no LDS round trip). Outputs use non-temporal b128
// stores so the write-once 205MB result doesn't evict the 256MB feature
// table from the 192MB L2 (rows are re-gathered ~4.4x on average).

typedef float v4f __attribute__((ext_vector_type(4)));

#define WAVES_PER_BLOCK 4
#define BLOCK_THREADS   (WAVES_PER_BLOCK * 32)
#define K_NEIGH         10
#define DIM             128                    // floats per feature row
#define ROW_BYTES       (DIM * 4)              // 512 B
#define ROWS_PER_B      (2 * K_NEIGH)          // 20 async-staged rows

__global__ __launch_bounds__(BLOCK_THREADS)
void MeanAggregator_gather_kernel(const float* __restrict__ features,
                                  const int*   __restrict__ nodes,
                                  const int*   __restrict__ neigh_idx,
                                  const int*   __restrict__ perm,
                                  float*       __restrict__ out,
                                  int B)
{
    // Per-wave LDS staging: 20 rows x 32 lanes x 16B = 10240 B per wave.
    __shared__ v4f smem[WAVES_PER_BLOCK][ROWS_PER_B][32];

    const int lane = threadIdx.x & 31;
    const int wave = threadIdx.x >> 5;
    const int b    = blockIdx.x * WAVES_PER_BLOCK + wave;
    if (b >= B) return;   // wave-uniform branch; EXEC stays all-ones inside

    // Lanes 0..K-1 fetch this batch element's neighbor indices in parallel,
    // then the permuted (double-indirect) indices.
    int ni = 0, pi = 0;
    if (lane < K_NEIGH) {
        ni = neigh_idx[b * K_NEIGH + lane];
        pi = perm[ni];
    }
    const int node  = nodes[b];
    const int pnode = perm[node];

    const uint64_t fbase     = (uint64_t)(uintptr_t)features;
    const uint32_t lane_byte = (uint32_t)lane * 16u;           // shared voffset
    const uint32_t lds_lane  = (uint32_t)(uintptr_t)&smem[wave][0][lane];

    // ---- Issue 20 async row gathers (one instruction = one 512B row:
    // 32 lanes x b128), global -> LDS, SADDR form. Tracked with ASYNCcnt.
#pragma unroll
    for (int k = 0; k < K_NEIGH; ++k) {
        const int idx  = __builtin_amdgcn_readlane(ni, k);     // uniform
        const int pidx = __builtin_amdgcn_readlane(pi, k);     // uniform
        const uint64_t sa  = fbase + (uint64_t)(uint32_t)idx  * ROW_BYTES;
        const uint64_t spa = fbase + (uint64_t)(uint32_t)pidx * ROW_BYTES;
        const uint32_t la  = lds_lane + (uint32_t)(k * ROW_BYTES);
        const uint32_t lpa = lds_lane + (uint32_t)((K_NEIGH + k) * ROW_BYTES);
        asm volatile("global_load_async_to_lds_b128 %0, %1, %2"
                     :: "v"(la),  "v"(lane_byte), "s"(sa)  : "memory");
        asm volatile("global_load_async_to_lds_b128 %0, %1, %2"
                     :: "v"(lpa), "v"(lane_byte), "s"(spa) : "memory");
    }

    // Skip rows: plain b128 loads straight to VGPRs, overlapped with the
    // async copies (no LDS round trip needed for a pure row copy).
    const v4f s0 = *(const v4f*)(fbase + (uint64_t)(uint32_t)node  * ROW_BYTES + lane_byte);
    const v4f s1 = *(const v4f*)(fbase + (uint64_t)(uint32_t)pnode * ROW_BYTES + lane_byte);

    // Drain this wave's async copies before touching the LDS staging area.
    asm volatile("s_wait_asynccnt 0" ::: "memory");

    // ---- Reduce out of LDS (ds_load_b128 per row slice). ----
    v4f acc0 = {0.f, 0.f, 0.f, 0.f};
    v4f acc1 = {0.f, 0.f, 0.f, 0.f};
#pragma unroll
    for (int k = 0; k < K_NEIGH; ++k) {
        acc0 += smem[wave][k][lane];
        acc1 += smem[wave][K_NEIGH + k][lane];
    }
    const float inv_k = 1.0f / (float)K_NEIGH;
    acc0 *= inv_k;
    acc1 *= inv_k;

    // ---- Non-temporal b128 stores; output tuple concatenated flat in
    // reference return order: to, shuf_to, skip, shuf_skip.
    const size_t BD  = (size_t)B * DIM;
    const size_t row = (size_t)b * DIM;
    __builtin_nontemporal_store(acc0, (v4f*)(out          + row) + lane);
    __builtin_nontemporal_store(acc1, (v4f*)(out + 1 * BD + row) + lane);
    __builtin_nontemporal_store(s0,   (v4f*)(out + 2 * BD + row) + lane);
    __builtin_nontemporal_store(s1,   (v4f*)(out + 3 * BD + row) + lane);
}

extern "C" void kernel_launch(void* const* d_in, const int* in_sizes, int n_in,
                              void* d_out, int out_size, void* d_ws, size_t ws_size,
                              hipStream_t stream) {
    // setup_inputs order: features [N*D] f32, nodes [B] i32,
    //                     neigh_idx [B*K] i32, perm [N] i32
    const float* features  = (const float*)d_in[0];
    const int*   nodes     = (const int*)  d_in[1];
    const int*   neigh_idx = (const int*)  d_in[2];
    const int*   perm      = (const int*)  d_in[3];
    float*       out       = (float*)d_out;

    const int B = in_sizes[1];
    const int grid = (B + WAVES_PER_BLOCK - 1) / WAVES_PER_BLOCK;

    MeanAggregator_gather_kernel<<<grid, BLOCK_THREADS, 0, stream>>>(
        features, nodes, neigh_idx, perm, out, B);
}